// TaskAlignedAssigner_75471165325891
// MI455X (gfx1250) — compile-verified
//
#include <hip/hip_runtime.h>
#include <hip/hip_bf16.h>
#include <stdint.h>

// Problem constants (from reference)
#define BS    32
#define A_N   8400
#define M_N   32
#define C_N   80
#define K_TOP 13
#define EPS   1e-9f

typedef float v2f __attribute__((ext_vector_type(2)));
typedef float v8f __attribute__((ext_vector_type(8)));

__device__ __forceinline__ float iou_xyxy(float g0, float g1, float g2, float g3,
                                          float p0, float p1, float p2, float p3) {
  float ltx = fmaxf(g0, p0), lty = fmaxf(g1, p1);
  float rbx = fminf(g2, p2), rby = fminf(g3, p3);
  float w = fmaxf(rbx - ltx, 0.f), h = fmaxf(rby - lty, 0.f);
  float inter = w * h;
  float ag = (g2 - g0) * (g3 - g1);
  float ap = (p2 - p0) * (p3 - p1);
  return inter / (ag + ap - inter + EPS);
}

// -------------------------------------------------------------------------
// K1: one wave per (b, 16-anchor tile). bbox_scores via WMMA f32 16x16x4
// (one-hot(labels) x score^T — exact gather in f32), fused with IoU and
// align = score * iou^6. Writes ovl[b,M,A] and alg[b,M,A] (raw align).
// -------------------------------------------------------------------------
__global__ __launch_bounds__(32)
void k1_metrics(const float* __restrict__ pd, const float* __restrict__ sc,
                const int* __restrict__ lab, const float* __restrict__ gt,
                float* __restrict__ ovl, float* __restrict__ alg) {
  const int t = blockIdx.x;      // anchor tile: 0..524
  const int b = blockIdx.y;      // batch
  const int lane = threadIdx.x;  // 0..31 (wave32)
  const int a0 = t * 16;

  __shared__ int   slab[M_N];
  __shared__ float sgt[M_N * 4];
  __shared__ float spd[16 * 4];

  slab[lane] = lab[b * M_N + lane];
  {
    const float* g = gt + (size_t)(b * M_N + lane) * 4;
    sgt[lane * 4 + 0] = g[0]; sgt[lane * 4 + 1] = g[1];
    sgt[lane * 4 + 2] = g[2]; sgt[lane * 4 + 3] = g[3];
  }
  if (lane < 16) {
    const float* p = pd + (size_t)(b * A_N + a0 + lane) * 4;
    spd[lane * 4 + 0] = p[0]; spd[lane * 4 + 1] = p[1];
    spd[lane * 4 + 2] = p[2]; spd[lane * 4 + 3] = p[3];
  }
  __syncthreads();

  const int half = lane >> 4;     // 0: lanes 0-15, 1: lanes 16-31
  const int q    = lane & 15;     // M (for A-matrix) and N (for B-matrix)
  const int kb   = half * 2;      // K pair held by this half-wave
  const int la   = slab[q];       // label of GT m=q      (low accumulator)
  const int lb   = slab[q + 16];  // label of GT m=q+16   (high accumulator)
  const float* srow = sc + ((size_t)b * A_N + a0 + q) * C_N;  // score row of anchor a0+q

  v8f clo = {0.f, 0.f, 0.f, 0.f, 0.f, 0.f, 0.f, 0.f};  // bbox_scores m=0..15
  v8f chi = {0.f, 0.f, 0.f, 0.f, 0.f, 0.f, 0.f, 0.f};  // bbox_scores m=16..31

  for (int k0 = 0; k0 < C_N; k0 += 4) {
    v2f av  = { (la == k0 + kb)     ? 1.f : 0.f,
                (la == k0 + kb + 1) ? 1.f : 0.f };
    v2f av2 = { (lb == k0 + kb)     ? 1.f : 0.f,
                (lb == k0 + kb + 1) ? 1.f : 0.f };
    v2f bv  = { srow[k0 + kb], srow[k0 + kb + 1] };
    clo = __builtin_amdgcn_wmma_f32_16x16x4_f32(false, av,  false, bv, (short)0, clo, false, false);
    chi = __builtin_amdgcn_wmma_f32_16x16x4_f32(false, av2, false, bv, (short)0, chi, false, false);
  }

  // C/D layout: VGPR r, lanes0-15 -> M=r, N=lane; lanes16-31 -> M=8+r, N=lane-16
  const int a = a0 + q;
  const float p0 = spd[q * 4 + 0], p1 = spd[q * 4 + 1];
  const float p2 = spd[q * 4 + 2], p3 = spd[q * 4 + 3];

#pragma unroll
  for (int r = 0; r < 8; ++r) {
    int mlo = r + 8 * half;
    {
      float g0 = sgt[mlo * 4], g1 = sgt[mlo * 4 + 1], g2 = sgt[mlo * 4 + 2], g3 = sgt[mlo * 4 + 3];
      float i1 = iou_xyxy(g0, g1, g2, g3, p0, p1, p2, p3);
      size_t idx = (size_t)(b * M_N + mlo) * A_N + a;
      float i2 = i1 * i1;
      ovl[idx] = i1;
      alg[idx] = clo[r] * i2 * i2 * i2;
    }
    int mhi = mlo + 16;
    {
      float g0 = sgt[mhi * 4], g1 = sgt[mhi * 4 + 1], g2 = sgt[mhi * 4 + 2], g3 = sgt[mhi * 4 + 3];
      float i1 = iou_xyxy(g0, g1, g2, g3, p0, p1, p2, p3);
      size_t idx = (size_t)(b * M_N + mhi) * A_N + a;
      float i2 = i1 * i1;
      ovl[idx] = i1;
      alg[idx] = chi[r] * i2 * i2 * i2;
    }
  }
}

// -------------------------------------------------------------------------
// K2: per-(b,m) top-13 over metrics = align * mask_in_gts.
// Row cached in LDS; 13 iterative block argmaxes with lax.top_k-compatible
// tie-break (value desc, index asc) via a single u64 key (values >= 0).
// Also zero-initializes the per-GT max accumulators.
// -------------------------------------------------------------------------
__global__ __launch_bounds__(256)
void k2_topk(const float* __restrict__ alg, const float* __restrict__ anch,
             const float* __restrict__ gt, int* __restrict__ topk,
             float* __restrict__ posal, float* __restrict__ posov) {
  const int row = blockIdx.x;       // b*M_N + m
  const int tid = threadIdx.x;
  __shared__ float sv[A_N];
  __shared__ unsigned long long sr[256];

  const float* g = gt + (size_t)row * 4;
  const float g0 = g[0], g1 = g[1], g2 = g[2], g3 = g[3];
  const float* arow = alg + (size_t)row * A_N;

  for (int a = tid; a < A_N; a += 256) {
    float ax = anch[2 * a], ay = anch[2 * a + 1];
    float d = fminf(fminf(ax - g0, ay - g1), fminf(g2 - ax, g3 - ay));
    sv[a] = (d > EPS) ? arow[a] : 0.0f;
  }
  if (tid == 0) { posal[row] = 0.f; posov[row] = 0.f; }
  __syncthreads();

  for (int it = 0; it < K_TOP; ++it) {
    unsigned long long best = 0ull;
    for (int a = tid; a < A_N; a += 256) {
      float v = sv[a];
      if (v >= 0.f) {  // skip -1 sentinels of already-picked entries
        unsigned long long key =
            ((unsigned long long)__float_as_uint(v) << 32) |
            (unsigned long long)(0xFFFFFFFFu - (unsigned)a);
        if (key > best) best = key;
      }
    }
    sr[tid] = best;
    __syncthreads();
    for (int s = 128; s > 0; s >>= 1) {
      if (tid < s) { if (sr[tid + s] > sr[tid]) sr[tid] = sr[tid + s]; }
      __syncthreads();
    }
    if (tid == 0) {
      unsigned idx = 0xFFFFFFFFu - (unsigned)(sr[0] & 0xFFFFFFFFull);
      topk[row * K_TOP + it] = (int)idx;
      sv[idx] = -1.0f;
    }
    __syncthreads();
  }
}

// -------------------------------------------------------------------------
// K3: per-anchor resolve. Final mask_pos has <=1 GT per anchor, so store
// (tgt, fg, align@tgt); atomic-max per-GT normalization stats (>=0 floats
// compare correctly as ints).
// -------------------------------------------------------------------------
__global__ __launch_bounds__(256)
void k3_resolve(const int* __restrict__ topk, const float* __restrict__ gt,
                const float* __restrict__ mgt, const float* __restrict__ anch,
                const float* __restrict__ ovl, const float* __restrict__ alg,
                int* __restrict__ tgti, int* __restrict__ fgb,
                float* __restrict__ alat,
                float* __restrict__ posal, float* __restrict__ posov) {
  const int b = blockIdx.y;
  const int tid = threadIdx.x;
  const int a = blockIdx.x * 256 + tid;

  __shared__ int   sid[M_N * K_TOP];
  __shared__ float sgt[M_N * 4];
  __shared__ float smk[M_N];

  for (int i = tid; i < M_N * K_TOP; i += 256) sid[i] = topk[b * M_N * K_TOP + i];
  for (int i = tid; i < M_N * 4; i += 256)     sgt[i] = gt[(size_t)b * M_N * 4 + i];
  if (tid < M_N) smk[tid] = mgt[b * M_N + tid];
  __syncthreads();

  if (a >= A_N) return;

  const float ax = anch[2 * a], ay = anch[2 * a + 1];
  int cnt = 0, first = -1;
  for (int m = 0; m < M_N; ++m) {
    if (smk[m] > 0.f) {
      float g0 = sgt[m * 4], g1 = sgt[m * 4 + 1], g2 = sgt[m * 4 + 2], g3 = sgt[m * 4 + 3];
      float d = fminf(fminf(ax - g0, ay - g1), fminf(g2 - ax, g3 - ay));
      if (d > EPS) {
        bool sel = false;
#pragma unroll
        for (int k = 0; k < K_TOP; ++k) sel |= (sid[m * K_TOP + k] == a);
        if (sel) { cnt++; if (first < 0) first = m; }
      }
    }
  }

  // argmax over all m of overlaps[b,:,a] (first occurrence of max)
  const float* orow = ovl + (size_t)b * M_N * A_N + a;
  float mv = orow[0]; int mm = 0;
  for (int m = 1; m < M_N; ++m) {
    float v = orow[(size_t)m * A_N];
    if (v > mv) { mv = v; mm = m; }
  }

  int tgt, fg;
  if (cnt > 1)      { tgt = mm;    fg = 1; }
  else if (cnt == 1){ tgt = first; fg = 1; }
  else              { tgt = 0;     fg = 0; }

  float av = 0.f;
  if (fg) {
    size_t idx = (size_t)(b * M_N + tgt) * A_N + a;
    av = alg[idx];
    float ov = ovl[idx];
    atomicMax((int*)&posal[b * M_N + tgt], __float_as_int(av));
    atomicMax((int*)&posov[b * M_N + tgt], __float_as_int(ov));
  }
  tgti[(size_t)b * A_N + a] = tgt;
  fgb [(size_t)b * A_N + a] = fg;
  alat[(size_t)b * A_N + a] = av;
}

// -------------------------------------------------------------------------
// K4: final outputs. labels/boxes/fg coalesced per anchor; scores written
// with a 256x80 index loop so consecutive lanes hit consecutive classes.
// -------------------------------------------------------------------------
__global__ __launch_bounds__(256)
void k4_outputs(const int* __restrict__ tgti, const int* __restrict__ fgb,
                const float* __restrict__ alat, const float* __restrict__ posal,
                const float* __restrict__ posov, const int* __restrict__ lab,
                const float* __restrict__ gt,
                float* __restrict__ out_lab, float* __restrict__ out_box,
                float* __restrict__ out_sc, float* __restrict__ out_fg) {
  const int b = blockIdx.y, tid = threadIdx.x;
  const int a0 = blockIdx.x * 256;
  const int a = a0 + tid;

  __shared__ float spa[M_N], spo[M_N], sgt[M_N * 4];
  __shared__ int   slab[M_N];
  __shared__ int   albl[256];
  __shared__ int   afg[256];
  __shared__ float anm[256];

  if (tid < M_N) {
    spa[tid] = posal[b * M_N + tid];
    spo[tid] = posov[b * M_N + tid];
    slab[tid] = lab[b * M_N + tid];
  }
  for (int i = tid; i < M_N * 4; i += 256) sgt[i] = gt[(size_t)b * M_N * 4 + i];
  __syncthreads();

  if (a < A_N) {
    int tgt  = tgti[(size_t)b * A_N + a];
    int fg   = fgb [(size_t)b * A_N + a];
    float av = alat[(size_t)b * A_N + a];
    float nrm = fg ? av * spo[tgt] / (spa[tgt] + EPS) : 0.f;
    int lbl = slab[tgt]; if (lbl < 0) lbl = 0;

    out_lab[(size_t)b * A_N + a] = (float)lbl;
    size_t bo = ((size_t)b * A_N + a) * 4;
    out_box[bo + 0] = sgt[tgt * 4 + 0];
    out_box[bo + 1] = sgt[tgt * 4 + 1];
    out_box[bo + 2] = sgt[tgt * 4 + 2];
    out_box[bo + 3] = sgt[tgt * 4 + 3];
    out_fg[(size_t)b * A_N + a] = fg ? 1.f : 0.f;

    albl[tid] = lbl; afg[tid] = fg; anm[tid] = nrm;
  } else {
    albl[tid] = -1; afg[tid] = 0; anm[tid] = 0.f;
  }
  __syncthreads();

  const size_t sbase = ((size_t)b * A_N + a0) * C_N;
  for (int idx = tid; idx < 256 * C_N; idx += 256) {
    int al = idx / C_N, c = idx % C_N;
    if (a0 + al < A_N) {
      float v = (afg[al] && c == albl[al]) ? anm[al] : 0.f;
      out_sc[sbase + (size_t)al * C_N + c] = v;
    }
  }
}

// -------------------------------------------------------------------------
extern "C" void kernel_launch(void* const* d_in, const int* in_sizes, int n_in,
                              void* d_out, int out_size, void* d_ws, size_t ws_size,
                              hipStream_t stream) {
  const float* pd   = (const float*)d_in[0];  // [32,8400,4]
  const float* sc   = (const float*)d_in[1];  // [32,8400,80]
  const float* anch = (const float*)d_in[2];  // [8400,2]
  const int*   lab  = (const int*)  d_in[3];  // [32,32,1]
  const float* gt   = (const float*)d_in[4];  // [32,32,4]
  const float* mgt  = (const float*)d_in[5];  // [32,32,1]
  float* out = (float*)d_out;

  const size_t NBM = (size_t)BS * M_N * A_N;
  char* ws = (char*)d_ws;
  float* ovl  = (float*)(ws);
  float* alg  = (float*)(ws + NBM * 4);
  int*   topk = (int*)  (ws + 2 * NBM * 4);
  size_t off = 2 * NBM * 4 + (size_t)BS * M_N * K_TOP * 4;
  int*   tgti = (int*)  (ws + off); off += (size_t)BS * A_N * 4;
  int*   fgb  = (int*)  (ws + off); off += (size_t)BS * A_N * 4;
  float* alat = (float*)(ws + off); off += (size_t)BS * A_N * 4;
  float* posal= (float*)(ws + off); off += (size_t)BS * M_N * 4;
  float* posov= (float*)(ws + off);

  float* out_lab = out;                             // [32,8400]
  float* out_box = out + (size_t)BS * A_N;          // [32,8400,4]
  float* out_sc  = out + (size_t)BS * A_N * 5;      // [32,8400,80]
  float* out_fg  = out + (size_t)BS * A_N * 85;     // [32,8400]

  k1_metrics<<<dim3(A_N / 16, BS), 32, 0, stream>>>(pd, sc, lab, gt, ovl, alg);
  k2_topk<<<dim3(BS * M_N), 256, 0, stream>>>(alg, anch, gt, topk, posal, posov);
  k3_resolve<<<dim3((A_N + 255) / 256, BS), 256, 0, stream>>>(
      topk, gt, mgt, anch, ovl, alg, tgti, fgb, alat, posal, posov);
  k4_outputs<<<dim3((A_N + 255) / 256, BS), 256, 0, stream>>>(
      tgti, fgb, alat, posal, posov, lab, gt, out_lab, out_box, out_sc, out_fg);
}